// NALUi1_22677427323454
// MI455X (gfx1250) — compile-verified
//
#include <hip/hip_runtime.h>
#include <hip/hip_bf16.h>
#include <math.h>

// NALU forward, fused as 4 fp32 WMMA GEMMs on gfx1250 (wave32).
// B=1024, I=O=512.

#define BB 1024
#define II 512
#define OO 512

typedef __attribute__((ext_vector_type(2))) float v2f;
typedef __attribute__((ext_vector_type(8))) float v8f;

__device__ __forceinline__ float sigmoidf_(float v) {
    return 1.0f / (1.0f + expf(-v));
}

// D = A(16x4 f32) * B(4x16 f32) + C(16x16 f32), full fp32 matrix op.
__device__ __forceinline__ v8f wmma_f32_k4(v2f a, v2f b, v8f c) {
#if defined(__gfx1250__) && __has_builtin(__builtin_amdgcn_wmma_f32_16x16x4_f32)
    // 8 args: (neg_a, A, neg_b, B, c_mod, C, reuse_a, reuse_b)
    return __builtin_amdgcn_wmma_f32_16x16x4_f32(
        /*neg_a=*/false, a, /*neg_b=*/false, b,
        /*c_mod=*/(short)0, c, /*reuse_a=*/false, /*reuse_b=*/false);
#else
    // Placeholder so compile still succeeds; histogram wmma==0 signals this path.
    c[0] += a[0] * b[0];
    return c;
#endif
}

// ---------------------------------------------------------------------------
// Prep 1: W[i,o] = tanh(w_hat)*sigmoid(m_hat)
//         L[i,o] = log|1 - 2|W[o,i]||   (transposed weight, square matrix)
//         P[i,o] = (1 - 2|W[o,i]| < 0) ? 1 : 0
// ---------------------------------------------------------------------------
__global__ __launch_bounds__(256) void nalu_prep_weights(
    const float* __restrict__ w_hat, const float* __restrict__ m_hat,
    float* __restrict__ W, float* __restrict__ L, float* __restrict__ P)
{
    int idx = blockIdx.x * blockDim.x + threadIdx.x;
    if (idx >= II * OO) return;
    int r = idx >> 9;      // row i
    int c = idx & 511;     // col o

    float w = tanhf(w_hat[idx]) * sigmoidf_(m_hat[idx]);
    W[idx] = w;

    int tidx = c * OO + r; // transposed element (I == O == 512)
    float wt = tanhf(w_hat[tidx]) * sigmoidf_(m_hat[tidx]);
    float t = 1.0f - 2.0f * fabsf(wt);
    L[idx] = logf(fmaxf(fabsf(t), 1e-38f));
    P[idx] = (t < 0.0f) ? 1.0f : 0.0f;
}

// ---------------------------------------------------------------------------
// Prep 2: XL[b,i] = log(max(|x|,1e-7)),  MK[b,i] = (x < 0) ? 1 : 0
// ---------------------------------------------------------------------------
__global__ __launch_bounds__(256) void nalu_prep_x(
    const float* __restrict__ x, float* __restrict__ XL, float* __restrict__ MK)
{
    int idx = blockIdx.x * blockDim.x + threadIdx.x;
    if (idx >= BB * II) return;
    float v = x[idx];
    XL[idx] = logf(fmaxf(fabsf(v), 1e-7f));
    MK[idx] = (v < 0.0f) ? 1.0f : 0.0f;
}

// ---------------------------------------------------------------------------
// Main: one 16x16 output tile per wave; four fp32 WMMA accumulators sharing
// the K loop (a = x@W, logm = XL@W, logms = MK@L, parity = MK@P).
//
// Fragment layouts (ISA 7.12.2, f32):
//   A 16x4 : lane L(0-15): M=L, VGPR0=K0,VGPR1=K1 ; lanes 16-31: K2,K3
//   B 4x16 : VGPR0: lanes0-15 row K0, lanes16-31 row K2; VGPR1: K1/K3
//   C 16x16: VGPR r: lanes0-15 M=r, lanes16-31 M=r+8; N=lane&15
// ---------------------------------------------------------------------------
__global__ __launch_bounds__(256) void nalu_wmma_kernel(
    const float* __restrict__ X,  const float* __restrict__ XL,
    const float* __restrict__ MK, const float* __restrict__ W,
    const float* __restrict__ L,  const float* __restrict__ P,
    const float* __restrict__ g,  float* __restrict__ out)
{
    const int lane  = threadIdx.x & 31;
    const int wave  = threadIdx.x >> 5;
    const int tile  = blockIdx.x * 8 + wave;   // 2048 tiles total
    const int mt    = tile >> 5;               // 64 row tiles
    const int nt    = tile & 31;               // 32 col tiles
    const int row   = mt * 16 + (lane & 15);   // A-fragment row
    const int col   = nt * 16 + (lane & 15);   // B/C-fragment column
    const int khalf = lane >> 4;               // 0: K={0,1}, 1: K={2,3}

    v8f acc_a = {}; // x @ W
    v8f acc_m = {}; // log|x| @ W
    v8f acc_s = {}; // mask @ L
    v8f acc_p = {}; // mask @ P  (integer-valued parity count)

    const float* xrow = X  + row * II;
    const float* lrow = XL + row * II;
    const float* krow = MK + row * II;

    #pragma unroll 4
    for (int k0 = 0; k0 < II; k0 += 4) {
        const int ka = k0 + khalf * 2;

        v2f ax = { xrow[ka], xrow[ka + 1] };
        v2f al = { lrow[ka], lrow[ka + 1] };
        v2f am = { krow[ka], krow[ka + 1] };

        v2f bw = { W[ka * OO + col], W[(ka + 1) * OO + col] };
        v2f bl = { L[ka * OO + col], L[(ka + 1) * OO + col] };
        v2f bp = { P[ka * OO + col], P[(ka + 1) * OO + col] };

        acc_a = wmma_f32_k4(ax, bw, acc_a);
        acc_m = wmma_f32_k4(al, bw, acc_m);
        acc_s = wmma_f32_k4(am, bl, acc_s);
        acc_p = wmma_f32_k4(am, bp, acc_p);
    }

    // Epilogue: gate, magnitude, sign-product reconstruction.
    const float gv  = sigmoidf_(g[col]);
    const float omg = 1.0f - gv;

    #pragma unroll
    for (int r = 0; r < 8; ++r) {
        const int orow = mt * 16 + r + khalf * 8;
        const float a  = acc_a[r];
        const float mm = expf(fminf(acc_m[r], 20.0f));
        float ms = expf(acc_s[r]);
        const int par = (int)(acc_p[r] + 0.5f);
        if (par & 1) ms = -ms;
        ms = fminf(fmaxf(ms, -1.0f), 1.0f);
        out[orow * OO + col] = gv * a + omg * mm * ms;
    }
}

// ---------------------------------------------------------------------------
extern "C" void kernel_launch(void* const* d_in, const int* in_sizes, int n_in,
                              void* d_out, int out_size, void* d_ws, size_t ws_size,
                              hipStream_t stream)
{
    const float* x     = (const float*)d_in[0];
    const float* w_hat = (const float*)d_in[1];
    const float* m_hat = (const float*)d_in[2];
    const float* g     = (const float*)d_in[3];
    float* out = (float*)d_out;

    // Workspace layout (floats): W | L | P | XL | MK  == 7 MB total.
    float* ws = (float*)d_ws;
    float* W  = ws;
    float* L  = ws + (size_t)II * OO;
    float* P  = ws + (size_t)2 * II * OO;
    float* XL = ws + (size_t)3 * II * OO;
    float* MK = XL + (size_t)BB * II;

    nalu_prep_weights<<<(II * OO + 255) / 256, 256, 0, stream>>>(w_hat, m_hat, W, L, P);
    nalu_prep_x<<<(BB * II + 255) / 256, 256, 0, stream>>>(x, XL, MK);

    // 2048 tiles, 8 waves (256 threads) per block -> 256 blocks.
    nalu_wmma_kernel<<<256, 256, 0, stream>>>(x, XL, MK, W, L, P, g, out);
}